// NGCF_implicit_19688130085762
// MI455X (gfx1250) — compile-verified
//
#include <hip/hip_runtime.h>

#define NUSERS 100000
#define NITEMS 200000
#define NNODE  (NUSERS + NITEMS)      // 300000
#define DIM    64
#define NLAY   3
#define NNZE   4000000
#define BATCH  4096

typedef __attribute__((ext_vector_type(2))) float v2f;
typedef __attribute__((ext_vector_type(8))) float v8f;

// ---------------------------------------------------------------------------
// ego0 = concat(user_emb, item_emb)   (float4 per thread, fully coalesced)
// ---------------------------------------------------------------------------
__global__ void ngcf_init_ego(const float* __restrict__ ue,
                              const float* __restrict__ ie,
                              float* __restrict__ ego) {
  int t = blockIdx.x * blockDim.x + threadIdx.x;       // NNODE*16 threads
  if (t >= NNODE * (DIM / 4)) return;
  int node = t >> 4;
  int c4   = (t & 15) << 2;
  const float* src = (node < NUSERS) ? (ue + (size_t)node * DIM + c4)
                                     : (ie + (size_t)(node - NUSERS) * DIM + c4);
  *(float4*)(ego + (size_t)node * DIM + c4) = *(const float4*)src;
}

// ---------------------------------------------------------------------------
// side[r,:] += vals[e] * ego[c,:]   — 16 threads per edge, float4 each,
// fp32 global atomics (side is L2-resident: 76.8MB < 192MB L2).
// ---------------------------------------------------------------------------
__global__ void ngcf_scatter(const float* __restrict__ vals,
                             const int*   __restrict__ rows,
                             const int*   __restrict__ cols,
                             const float* __restrict__ ego,
                             float* __restrict__ side) {
  long t = (long)blockIdx.x * blockDim.x + threadIdx.x;  // NNZE*16 threads
  long e = t >> 4;
  if (e >= NNZE) return;
  int c4 = (int)(t & 15) << 2;
  float v = vals[e];
  int r = rows[e];
  int c = cols[e];
  float4 x = *(const float4*)(ego + (size_t)c * DIM + c4);
  float* dst = side + (size_t)r * DIM + c4;
  __hip_atomic_fetch_add(dst + 0, v * x.x, __ATOMIC_RELAXED, __HIP_MEMORY_SCOPE_AGENT);
  __hip_atomic_fetch_add(dst + 1, v * x.y, __ATOMIC_RELAXED, __HIP_MEMORY_SCOPE_AGENT);
  __hip_atomic_fetch_add(dst + 2, v * x.z, __ATOMIC_RELAXED, __HIP_MEMORY_SCOPE_AGENT);
  __hip_atomic_fetch_add(dst + 3, v * x.w, __ATOMIC_RELAXED, __HIP_MEMORY_SCOPE_AGENT);
}

// ---------------------------------------------------------------------------
// Fused layer: out = leaky_relu(side@W1 + b1 + (ego*side)@W2 + b2)
//   ego_next = out (unnormalized, carried to next layer)
//   norm_out = out / max(||row||, 1e-12)
// Block = 128 threads (4 waves). Block handles 16 rows; wave w handles the
// 16-column tile n0 = 16*w. K=64 -> 16 steps of V_WMMA_F32_16X16X4_F32 per
// GEMM; both GEMMs accumulate into one v8f.
//
// WMMA f32 16x16x4 operand layout (ISA 7.12.2):
//   A 16x4 : v0 = A[m][k0+kh], v1 = A[m][k0+kh+1], m = lane&15, kh = 2*(lane>>4)
//   B 4x16 : v0 = B[k0+kh][n], v1 = B[k0+kh+1][n], n = lane&15
//   C 16x16: vgpr j = row (j + 8*(lane>>4)), col = lane&15 (within tile)
// ---------------------------------------------------------------------------
__global__ __launch_bounds__(128)
void ngcf_layer(const float* __restrict__ side,
                const float* __restrict__ ego,
                const float* __restrict__ W1, const float* __restrict__ b1,
                const float* __restrict__ W2, const float* __restrict__ b2,
                float* __restrict__ ego_next,
                float* __restrict__ norm_out) {
  __shared__ float snorm[16];
  const int lane  = threadIdx.x & 31;
  const int w     = threadIdx.x >> 5;          // 0..3: column tile
  const int n0    = w << 4;
  const int m     = lane & 15;                 // A row within tile
  const int kh    = (lane >> 4) << 1;          // K-pair select (0 or 2)
  const int col   = n0 + (lane & 15);          // output column (0..63)
  const long m0   = (long)blockIdx.x << 4;     // first row of tile

  if (threadIdx.x < 16) snorm[threadIdx.x] = 0.0f;
  __syncthreads();

  const float* sA = side + (m0 + m) * DIM;
  const float* eA = ego  + (m0 + m) * DIM;

  v8f acc = {};
#pragma unroll
  for (int kk = 0; kk < 16; ++kk) {
    const int k0 = kk << 2;
    float2 s2 = *(const float2*)(sA + k0 + kh);
    float2 e2 = *(const float2*)(eA + k0 + kh);
    v2f a_sum, a_bi, bw1, bw2;
    a_sum.x = s2.x;        a_sum.y = s2.y;
    a_bi.x  = s2.x * e2.x; a_bi.y  = s2.y * e2.y;
    bw1.x = W1[(k0 + kh) * DIM + col];
    bw1.y = W1[(k0 + kh + 1) * DIM + col];
    bw2.x = W2[(k0 + kh) * DIM + col];
    bw2.y = W2[(k0 + kh + 1) * DIM + col];
    acc = __builtin_amdgcn_wmma_f32_16x16x4_f32(false, a_sum, false, bw1,
                                                (short)0, acc, false, false);
    acc = __builtin_amdgcn_wmma_f32_16x16x4_f32(false, a_bi, false, bw2,
                                                (short)0, acc, false, false);
  }

  const float bias = b1[col] + b2[col];
  const int   rsel = (lane >> 4) << 3;         // +0 or +8 row offset

  float x[8];
#pragma unroll
  for (int j = 0; j < 8; ++j) {
    float v = acc[j] + bias;
    x[j] = (v > 0.0f) ? v : 0.2f * v;          // leaky_relu(0.2)
    // partial row-norm over this wave's 16 columns
    float sq = x[j] * x[j];
    sq += __shfl_xor(sq, 1, 32);
    sq += __shfl_xor(sq, 2, 32);
    sq += __shfl_xor(sq, 4, 32);
    sq += __shfl_xor(sq, 8, 32);
    if ((lane & 15) == 0) atomicAdd(&snorm[j + rsel], sq);   // ds_add_f32
    // unnormalized activation -> next layer input
    ego_next[(m0 + j + rsel) * DIM + col] = x[j];
  }
  __syncthreads();

#pragma unroll
  for (int j = 0; j < 8; ++j) {
    float nrm = sqrtf(snorm[j + rsel]);
    float inv = 1.0f / fmaxf(nrm, 1e-12f);
    norm_out[(m0 + j + rsel) * DIM + col] = x[j] * inv;
  }
}

// ---------------------------------------------------------------------------
// Gather: out[3][BATCH][256] ; cols 0:64 from raw embeddings (all_embs[0]),
// cols 64*(k+1) from normalized layer-k output.
// ---------------------------------------------------------------------------
__global__ void ngcf_gather(const float* __restrict__ ue,
                            const float* __restrict__ ie,
                            const int* __restrict__ users,
                            const int* __restrict__ pos,
                            const int* __restrict__ neg,
                            const float* __restrict__ n1,
                            const float* __restrict__ n2,
                            const float* __restrict__ n3,
                            float* __restrict__ out) {
  int t = blockIdx.x * blockDim.x + threadIdx.x;     // 3*BATCH*64 threads
  if (t >= 3 * BATCH * 64) return;
  int which = t / (BATCH * 64);
  int b     = (t >> 6) & (BATCH - 1);
  int c4    = (t & 63) << 2;                         // column chunk (0..252)
  int idx   = (which == 0) ? users[b] : ((which == 1) ? pos[b] : neg[b]);
  long node = (which == 0) ? (long)idx : (long)NUSERS + idx;

  float4 v;
  if (c4 < 64) {
    const float* src = (which == 0) ? (ue + (size_t)idx * DIM + c4)
                                    : (ie + (size_t)idx * DIM + c4);
    v = *(const float4*)src;
  } else if (c4 < 128) {
    v = *(const float4*)(n1 + node * DIM + (c4 - 64));
  } else if (c4 < 192) {
    v = *(const float4*)(n2 + node * DIM + (c4 - 128));
  } else {
    v = *(const float4*)(n3 + node * DIM + (c4 - 192));
  }
  *(float4*)(out + (size_t)which * BATCH * 256 + (size_t)b * 256 + c4) = v;
}

// ---------------------------------------------------------------------------
extern "C" void kernel_launch(void* const* d_in, const int* in_sizes, int n_in,
                              void* d_out, int out_size, void* d_ws, size_t ws_size,
                              hipStream_t stream) {
  const float* user_emb = (const float*)d_in[0];
  const float* item_emb = (const float*)d_in[1];
  const float* W1       = (const float*)d_in[2];   // (L,64,64)
  const float* b1       = (const float*)d_in[3];   // (L,1,64)
  const float* W2       = (const float*)d_in[4];
  const float* b2       = (const float*)d_in[5];
  const float* adj_vals = (const float*)d_in[6];
  const int*   adj_rows = (const int*)d_in[7];
  const int*   adj_cols = (const int*)d_in[8];
  const int*   users    = (const int*)d_in[9];
  const int*   pos      = (const int*)d_in[10];
  const int*   neg      = (const int*)d_in[11];
  float* out = (float*)d_out;

  // Workspace layout (floats): ego_a | ego_b | side | norm1 | norm2 | norm3
  const size_t stride = (size_t)NNODE * DIM;
  float* wsf   = (float*)d_ws;
  float* ego_a = wsf;
  float* ego_b = wsf + stride;
  float* side  = wsf + 2 * stride;
  float* norms[NLAY] = { wsf + 3 * stride, wsf + 4 * stride, wsf + 5 * stride };

  // ego0 = concat(user_emb, item_emb)
  {
    int total = NNODE * (DIM / 4);
    ngcf_init_ego<<<(total + 255) / 256, 256, 0, stream>>>(user_emb, item_emb, ego_a);
  }

  float* cur = ego_a;
  float* nxt = ego_b;
  for (int k = 0; k < NLAY; ++k) {
    hipMemsetAsync(side, 0, stride * sizeof(float), stream);

    long total = (long)NNZE * 16;
    ngcf_scatter<<<(int)((total + 255) / 256), 256, 0, stream>>>(
        adj_vals, adj_rows, adj_cols, cur, side);

    ngcf_layer<<<NNODE / 16, 128, 0, stream>>>(
        side, cur,
        W1 + (size_t)k * DIM * DIM, b1 + (size_t)k * DIM,
        W2 + (size_t)k * DIM * DIM, b2 + (size_t)k * DIM,
        nxt, norms[k]);

    float* t = cur; cur = nxt; nxt = t;
  }

  {
    int total = 3 * BATCH * 64;
    ngcf_gather<<<(total + 255) / 256, 256, 0, stream>>>(
        user_emb, item_emb, users, pos, neg,
        norms[0], norms[1], norms[2], out);
  }
}